// NeuralMemory_32220844655354
// MI455X (gfx1250) — compile-verified
//
#include <hip/hip_runtime.h>
#include <math.h>

typedef float v2f __attribute__((ext_vector_type(2)));
typedef float v4f __attribute__((ext_vector_type(4)));
typedef float v8f __attribute__((ext_vector_type(8)));

#define BB 16
#define TT 4096
#define DIN 64
#define DH 128
#define HH 4
#define NN 128
#define MM 64
// D_CTRL = HH*NN + DH = 640

#define EPSV 1e-8f

__device__ __forceinline__ float sigf(float x) { return 1.0f / (1.0f + __expf(-x)); }

__global__ __launch_bounds__(256, 1)
void ntm_persistent_kernel(const float* __restrict__ x,
                           const float* __restrict__ lstm_Wx,
                           const float* __restrict__ lstm_Wh,
                           const float* __restrict__ lstm_b,
                           const float* __restrict__ k_W, const float* __restrict__ k_b,
                           const float* __restrict__ e_W, const float* __restrict__ e_b,
                           const float* __restrict__ a_W, const float* __restrict__ a_b,
                           const float* __restrict__ r_W, const float* __restrict__ r_b,
                           const float* __restrict__ w_w0,
                           const float* __restrict__ w_r0,
                           float* __restrict__ mem,      // workspace: BB*NN*MM floats
                           float* __restrict__ out)      // BB*HH*MM floats
{
    // ---- LDS state (one WGP owns the whole recurrence; ~252KB of 320KB) ----
    __shared__ float sh_x[BB * DIN];        // 1024
    __shared__ float sh_h[BB * DH];         // 2048
    __shared__ float sh_c[BB * DH];         // 2048
    __shared__ float sh_z[BB * 4 * DH];     // 8192
    __shared__ float sh_ww[BB * HH * NN];   // 8192  write weights  [b][h*128+n]
    __shared__ float sh_wr[BB * HH * NN];   // 8192  read weights
    __shared__ float sh_kear[BB * 1024];    // 16384 [b][mat*256 + h*64 + m], mat: 0=k 1=e 2=a 3=r
    __shared__ float sh_scw[BB * HH * NN];  // 8192  write-head attention logits
    __shared__ float sh_scr[BB * HH * NN];  // 8192  read-head attention logits
    __shared__ float sh_norm[BB * NN];      // 2048  memory row norms (+EPS)

    const int tid  = threadIdx.x;
    const int lane = tid & 31;
    const int wave = tid >> 5;
    const int hl   = lane >> 4;   // 0/1: which 16-lane half
    const int l16  = lane & 15;

    // hint the weight matrices into cache early (global_prefetch_b8)
    if (tid < 32) {
        __builtin_prefetch(lstm_Wx + lane * 1024, 0, 3);
        __builtin_prefetch(lstm_Wh + lane * 2048, 0, 3);
        __builtin_prefetch(k_W + lane * 5120, 0, 3);
        __builtin_prefetch(r_W + lane * 5120, 0, 3);
    }

    // per-lane LDS bases for WMMA A fragments (contiguous v2f pairs)
    const v2f* Ax = (const v2f*)(sh_x + l16 * DIN + 2 * hl);   // Ax[2*kk] = {k, k+1}
    const v2f* Ah = (const v2f*)(sh_h + l16 * DH + 2 * hl);
    const v2f* Aww = (const v2f*)(sh_ww + l16 * 512 + 2 * hl);
    const v2f* Awr = (const v2f*)(sh_wr + l16 * 512 + 2 * hl);

    // ---- init: mem = 0, load w_w0/w_r0, h = c = 0 ----
    for (int i = tid; i < BB * NN * MM; i += 256) mem[i] = 0.0f;
    for (int i = tid; i < BB * HH * NN; i += 256) { sh_ww[i] = w_w0[i]; sh_wr[i] = w_r0[i]; }
    for (int i = tid; i < BB * DH; i += 256) { sh_h[i] = 0.0f; sh_c[i] = 0.0f; }
    __syncthreads();

    for (int t = 0; t < TT; ++t) {
        // ================= LSTM step: z = [x_t | h] @ [Wx;Wh] + b =================
        for (int i = tid; i < BB * DIN; i += 256)
            sh_x[i] = x[((i >> 6) * TT + t) * DIN + (i & 63)];
        __syncthreads();

        // 32 N-tiles of 16 cols over 512; 4 tiles per wave; K = 192 -> 48 WMMA steps
        for (int nt = wave * 4; nt < wave * 4 + 4; ++nt) {
            const int n = nt * 16 + l16;
            const float bias = lstm_b[n];
            const float* Bx = lstm_Wx + 2 * hl * 512 + n;   // Bx[(4kk+j)*512]
            const float* Bh = lstm_Wh + 2 * hl * 512 + n;
            v8f acc;
#pragma unroll
            for (int r = 0; r < 8; ++r) acc[r] = bias;
#pragma unroll 4
            for (int kk = 0; kk < 16; ++kk) {               // K = 0..63 (x part)
                const v2f af = Ax[kk * 2];
                v2f bf; bf.x = Bx[(kk * 4) * 512]; bf.y = Bx[(kk * 4 + 1) * 512];
                acc = __builtin_amdgcn_wmma_f32_16x16x4_f32(false, af, false, bf,
                                                            (short)0, acc, false, false);
            }
#pragma unroll 4
            for (int kk = 0; kk < 32; ++kk) {               // K = 64..191 (h part)
                const v2f af = Ah[kk * 2];
                v2f bf; bf.x = Bh[(kk * 4) * 512]; bf.y = Bh[(kk * 4 + 1) * 512];
                acc = __builtin_amdgcn_wmma_f32_16x16x4_f32(false, af, false, bf,
                                                            (short)0, acc, false, false);
            }
#pragma unroll
            for (int r = 0; r < 8; ++r) sh_z[(r + 8 * hl) * 512 + n] = acc[r];
        }
        __syncthreads();

        // gates (keras order i,f,g,o)
#pragma unroll
        for (int it = 0; it < 8; ++it) {
            const int idx = tid * 8 + it;          // 0..2047
            const int b = idx >> 7, j = idx & 127;
            const float* zr = &sh_z[b * 512];
            const float zi = zr[j], zf = zr[128 + j], zg = zr[256 + j], zo = zr[384 + j];
            const float c = sigf(zf) * sh_c[idx] + sigf(zi) * tanhf(zg);
            sh_c[idx] = c;
            sh_h[idx] = sigf(zo) * tanhf(c);
        }
        __syncthreads();

        if (t >= TT - 1) break;   // memory loop consumes ctrl[0..T-2]

        // ================= NTM memory step (o_t = sh_h) =================
        // (a) memory row norms (pre-update memory)
        for (int row = tid; row < BB * NN; row += 256) {
            const v4f* mr = (const v4f*)(mem + row * MM);
            float s = 0.0f;
#pragma unroll
            for (int q = 0; q < MM / 4; ++q) {
                const v4f v = mr[q];
                s += v.x * v.x + v.y * v.y + v.z * v.z + v.w * v.w;
            }
            sh_norm[row] = sqrtf(s) + EPSV;
        }

        // (b) k/e/a/r = inp @ W + b : M=16, 1024 cols (4 matrices x 256), K=640 -> 160 steps
        for (int tt2 = wave * 8; tt2 < wave * 8 + 8; ++tt2) {
            const int mat = tt2 >> 4;                   // 0=k 1=e 2=a 3=r (wave-uniform)
            const int col = (tt2 & 15) * 16 + l16;      // 0..255
            const float* W  = (mat == 0) ? k_W : (mat == 1) ? e_W : (mat == 2) ? a_W : r_W;
            const float* bp = (mat == 0) ? k_b : (mat == 1) ? e_b : (mat == 2) ? a_b : r_b;
            const v2f* Aw = (mat == 3) ? Awr : Aww;     // inp_r uses w_r
            const float* Bw = W + 2 * hl * 256 + col;           // k = 0..511
            const float* Bh2 = W + (512 + 2 * hl) * 256 + col;  // k = 512..639
            const float bias = bp[col];
            v8f acc;
#pragma unroll
            for (int r = 0; r < 8; ++r) acc[r] = bias;
#pragma unroll 4
            for (int kk = 0; kk < 128; ++kk) {          // K = 0..511 (w_w / w_r part)
                const v2f af = Aw[kk * 2];
                v2f bf; bf.x = Bw[(kk * 4) * 256]; bf.y = Bw[(kk * 4 + 1) * 256];
                acc = __builtin_amdgcn_wmma_f32_16x16x4_f32(false, af, false, bf,
                                                            (short)0, acc, false, false);
            }
#pragma unroll 4
            for (int kk = 0; kk < 32; ++kk) {           // K = 512..639 (o_t part)
                const v2f af = Ah[kk * 2];
                v2f bf; bf.x = Bh2[(kk * 4) * 256]; bf.y = Bh2[(kk * 4 + 1) * 256];
                acc = __builtin_amdgcn_wmma_f32_16x16x4_f32(false, af, false, bf,
                                                            (short)0, acc, false, false);
            }
#pragma unroll
            for (int r = 0; r < 8; ++r) {
                float v = acc[r];
                if (mat == 1 || mat == 2) v = sigf(v);   // e, a -> sigmoid
                sh_kear[(r + 8 * hl) * 1024 + mat * 256 + col] = v;
            }
        }
        __syncthreads();

        // (c) normalize kn / rn rows (64 rows each of length 64)
        if (tid < 128) {
            const int sel = tid >> 6;                 // 0: k, 1: r
            const int row = tid & 63;                 // b*4 + h
            const int b = row >> 2, h = row & 3;
            v4f* p = (v4f*)&sh_kear[b * 1024 + (sel ? 768 : 0) + h * 64];
            float s = 0.0f;
#pragma unroll
            for (int q = 0; q < MM / 4; ++q) {
                const v4f v = p[q];
                s += v.x * v.x + v.y * v.y + v.z * v.z + v.w * v.w;
            }
            const float inv = 1.0f / (sqrtf(s) + EPSV);
#pragma unroll
            for (int q = 0; q < MM / 4; ++q) p[q] *= inv;
        }
        __syncthreads();

        // (d) attention logits: dot(kn, mem_row)/norm, both heads
        for (int idx = tid; idx < 2 * BB * HH * NN; idx += 256) {
            const int sel = idx >> 13;
            const int rem = idx & 8191;
            const int b = rem >> 9, h = (rem >> 7) & 3, n = rem & 127;
            const v4f* kv = (const v4f*)&sh_kear[b * 1024 + (sel ? 768 : 0) + h * 64];
            const v4f* mr = (const v4f*)(mem + (b * NN + n) * MM);
            float s = 0.0f;
#pragma unroll
            for (int q = 0; q < MM / 4; ++q) {
                const v4f a4 = kv[q], m4 = mr[q];
                s += a4.x * m4.x + a4.y * m4.y + a4.z * m4.z + a4.w * m4.w;
            }
            s /= sh_norm[b * NN + n];
            (sel ? sh_scr : sh_scw)[rem] = s;
        }
        __syncthreads();

        // (e) softmax over n=128 per (head,b,h); results become new w_w / w_r
        if (tid < 128) {
            const int sel = tid >> 6;
            const int row = tid & 63;
            const float* src = (sel ? sh_scr : sh_scw) + row * 128;
            float*       dst = (sel ? sh_wr : sh_ww) + row * 128;
            float mx = -INFINITY;
            for (int n = 0; n < NN; ++n) mx = fmaxf(mx, src[n]);
            float s = 0.0f;
            for (int n = 0; n < NN; ++n) { const float ev = __expf(src[n] - mx); dst[n] = ev; s += ev; }
            const float inv = 1.0f / s;
            for (int n = 0; n < NN; ++n) dst[n] *= inv;
        }
        __syncthreads();

        // (f) final read (pre-update memory) -- only the last one is returned
        if (t == TT - 2) {
            for (int i4 = tid; i4 < BB * HH * MM / 4; i4 += 256) {   // 1024 v4 outputs
                const int idx = i4 * 4;
                const int b = idx >> 8, h = (idx >> 6) & 3, m0 = idx & 63;
                const float* wr = &sh_wr[b * 512 + h * 128];
                v4f s; s.x = s.y = s.z = s.w = 0.0f;
                for (int n = 0; n < NN; ++n) {
                    const float w = wr[n];
                    const v4f m4 = *(const v4f*)(mem + (b * NN + n) * MM + m0);
                    s.x += w * m4.x; s.y += w * m4.y; s.z += w * m4.z; s.w += w * m4.w;
                }
                *(v4f*)(out + idx) = s;
            }
            __syncthreads();
        }

        // (g) memory update: mem = mem - mem*(mem*erase) + add, rank-4 over heads
        for (int i4 = tid; i4 < BB * NN * MM / 4; i4 += 256) {       // 32768 v4 elements
            const int idx = i4 * 4;
            const int b = idx >> 13, n = (idx >> 6) & 127, m0 = idx & 63;
            v4f er; er.x = er.y = er.z = er.w = 0.0f;
            v4f ad; ad.x = ad.y = ad.z = ad.w = 0.0f;
#pragma unroll
            for (int h = 0; h < HH; ++h) {
                const float w = sh_ww[b * 512 + h * 128 + n];
                const v4f e4 = *(const v4f*)&sh_kear[b * 1024 + 256 + h * 64 + m0];
                const v4f a4 = *(const v4f*)&sh_kear[b * 1024 + 512 + h * 64 + m0];
                er.x += w * e4.x; er.y += w * e4.y; er.z += w * e4.z; er.w += w * e4.w;
                ad.x += w * a4.x; ad.y += w * a4.y; ad.z += w * a4.z; ad.w += w * a4.w;
            }
            v4f mv = *(const v4f*)(mem + idx);
            mv.x = mv.x - mv.x * (mv.x * er.x) + ad.x;
            mv.y = mv.y - mv.y * (mv.y * er.y) + ad.y;
            mv.z = mv.z - mv.z * (mv.z * er.z) + ad.z;
            mv.w = mv.w - mv.w * (mv.w * er.w) + ad.w;
            *(v4f*)(mem + idx) = mv;
        }
        __syncthreads();
    }
}

extern "C" void kernel_launch(void* const* d_in, const int* in_sizes, int n_in,
                              void* d_out, int out_size, void* d_ws, size_t ws_size,
                              hipStream_t stream) {
    const float* x       = (const float*)d_in[0];
    const float* lstm_Wx = (const float*)d_in[1];
    const float* lstm_Wh = (const float*)d_in[2];
    const float* lstm_b  = (const float*)d_in[3];
    const float* k_W     = (const float*)d_in[4];
    const float* k_b     = (const float*)d_in[5];
    const float* e_W     = (const float*)d_in[6];
    const float* e_b     = (const float*)d_in[7];
    const float* a_W     = (const float*)d_in[8];
    const float* a_b     = (const float*)d_in[9];
    const float* r_W     = (const float*)d_in[10];
    const float* r_b     = (const float*)d_in[11];
    const float* w_w0    = (const float*)d_in[12];
    const float* w_r0    = (const float*)d_in[13];
    float* mem = (float*)d_ws;            // BB*NN*MM floats = 512 KB
    float* out = (float*)d_out;           // BB*HH*MM floats

    ntm_persistent_kernel<<<1, 256, 0, stream>>>(
        x, lstm_Wx, lstm_Wh, lstm_b, k_W, k_b, e_W, e_b, a_W, a_b, r_W, r_b,
        w_w0, w_r0, mem, out);
}